// GNN_22943715295836
// MI455X (gfx1250) — compile-verified
//
#include <hip/hip_runtime.h>
#include <math.h>

#define N_NODES  50000
#define N_EDGES  800000
#define N_GRAPHS 64
#define IN_CH    128
#define H        64      // H1 == H2 == 64
#define NCLS     10

typedef float v2f __attribute__((ext_vector_type(2)));
typedef float v8f __attribute__((ext_vector_type(8)));

// ---------------------------------------------------------------------------
// C[M x 64] = (RELU_IN ? relu(A) : A)[M x K] @ B[K x 64], fp32 WMMA 16x16x4.
// One 4-wave block per 16-row stripe; wave w owns the 16x16 tile at col w*16.
// A-operand layout (ISA 32-bit A 16x4): lane l<16 -> A[m0+l][k+0..1],
// lane 16+l -> A[m0+l][k+2..3].  B layout: VGPR0 = K rows {0,2}, VGPR1 = {1,3}
// striped over N across lanes.  C: VGPR r -> rows {m0+r, m0+r+8}.
// ---------------------------------------------------------------------------
template <int K, bool RELU_IN>
__global__ void gemm16_wmma(const float* __restrict__ A,
                            const float* __restrict__ B,
                            float* __restrict__ C) {
    const int lane = threadIdx.x & 31;
    const int wave = threadIdx.x >> 5;
    const int l    = lane & 15;
    const int half = lane >> 4;          // 0: K pair {0,1}, 1: K pair {2,3}
    const int m0   = blockIdx.x * 16;
    const int n0   = wave * 16;

    const float* arow = A + (size_t)(m0 + l) * K + 2 * half;   // 8B aligned
    const float* bcol = B + (size_t)(2 * half) * H + n0 + l;

    v8f acc = (v8f)0.0f;
#pragma unroll
    for (int k = 0; k < K; k += 4) {
        v2f a = *(const v2f*)(arow + k);                 // A[m][k+2h], A[m][k+2h+1]
        if (RELU_IN) {
            a.x = fmaxf(a.x, 0.0f);
            a.y = fmaxf(a.y, 0.0f);
        }
        v2f b;
        b.x = bcol[(size_t)k * H];                       // B[k+2h  ][n]
        b.y = bcol[(size_t)(k + 1) * H];                 // B[k+2h+1][n]
        acc = __builtin_amdgcn_wmma_f32_16x16x4_f32(
            /*neg_a=*/false, a, /*neg_b=*/false, b,
            /*c_mod=*/(short)0, acc, /*reuse_a=*/false, /*reuse_b=*/false);
    }

    float* crow = C + (size_t)(m0 + 8 * half) * H + n0 + l;
#pragma unroll
    for (int r = 0; r < 8; ++r)
        crow[(size_t)r * H] = acc[r];
}

// agg[node][f] = bias[f]  (fuses "+ b" of the GCN layer into the accumulator)
__global__ void init_bias(float* __restrict__ agg, const float* __restrict__ bias) {
    size_t i = (size_t)blockIdx.x * blockDim.x + threadIdx.x;
    if (i < (size_t)N_NODES * H) agg[i] = bias[i & (H - 1)];
}

// One wave per edge: agg[dst[e]][:] += h[src[e]][:]  (64 feats, 2 per lane).
// src[e]/dst[e] are wave-uniform -> scalar loads; atomics resolve in L2
// (both 12.8 MB buffers are L2-resident on the 192 MB L2).
__global__ void edge_scatter(const float* __restrict__ h,
                             const int* __restrict__ src,
                             const int* __restrict__ dst,
                             float* __restrict__ agg) {
    const int lane = threadIdx.x & 31;
    const int wave = threadIdx.x >> 5;
    const int e = blockIdx.x * (blockDim.x >> 5) + wave;
    if (e >= N_EDGES) return;
    const int s = src[e];
    const int d = dst[e];
    const float* hs = h + (size_t)s * H;
    float* ad = agg + (size_t)d * H;
    atomicAdd(&ad[lane],      hs[lane]);
    atomicAdd(&ad[lane + 32], hs[lane + 32]);
}

__global__ void zero_pool(float* __restrict__ pool, float* __restrict__ cnt) {
    int i = blockIdx.x * blockDim.x + threadIdx.x;
    if (i < N_GRAPHS * H) pool[i] = 0.0f;
    if (i < N_GRAPHS)     cnt[i]  = 0.0f;
}

// pool[batch[n]][f] += relu(agg[n][f]); counts via lane f==0
__global__ void pool_scatter(const float* __restrict__ agg,
                             const int* __restrict__ batch,
                             float* __restrict__ pool,
                             float* __restrict__ cnt) {
    size_t i = (size_t)blockIdx.x * blockDim.x + threadIdx.x;
    if (i >= (size_t)N_NODES * H) return;
    const int node = (int)(i >> 6);
    const int f    = (int)(i & 63);
    const int g    = batch[node];
    atomicAdd(&pool[g * H + f], fmaxf(agg[i], 0.0f));
    if (f == 0) atomicAdd(&cnt[g], 1.0f);
}

// Tiny head: one thread per graph. mean -> fc1+relu -> fc2 -> log_softmax.
__global__ void head_kernel(const float* __restrict__ pool,
                            const float* __restrict__ cnt,
                            const float* __restrict__ fc1_w,  // [64 x 32]
                            const float* __restrict__ fc1_b,  // [32]
                            const float* __restrict__ fc2_w,  // [32 x 10]
                            const float* __restrict__ fc2_b,  // [10]
                            float* __restrict__ out) {        // [64 x 10]
    const int g = threadIdx.x;
    if (g >= N_GRAPHS) return;
    const float inv = 1.0f / fmaxf(cnt[g], 1.0f);

    float p[H];
#pragma unroll
    for (int f = 0; f < H; ++f) p[f] = pool[g * H + f] * inv;

    float z1[32];
#pragma unroll
    for (int j = 0; j < 32; ++j) {
        float s = fc1_b[j];
#pragma unroll
        for (int f = 0; f < H; ++f) s += p[f] * fc1_w[f * 32 + j];
        z1[j] = fmaxf(s, 0.0f);
    }

    float z2[NCLS];
    float m = -INFINITY;
#pragma unroll
    for (int j = 0; j < NCLS; ++j) {
        float s = fc2_b[j];
#pragma unroll
        for (int i = 0; i < 32; ++i) s += z1[i] * fc2_w[i * NCLS + j];
        z2[j] = s;
        m = fmaxf(m, s);
    }
    float se = 0.0f;
#pragma unroll
    for (int j = 0; j < NCLS; ++j) se += expf(z2[j] - m);
    const float lse = m + logf(se);
#pragma unroll
    for (int j = 0; j < NCLS; ++j) out[g * NCLS + j] = z2[j] - lse;
}

extern "C" void kernel_launch(void* const* d_in, const int* in_sizes, int n_in,
                              void* d_out, int out_size, void* d_ws, size_t ws_size,
                              hipStream_t stream) {
    const float* x     = (const float*)d_in[0];
    const int*   eidx  = (const int*)  d_in[1];   // [2 x N_EDGES], row-major
    const int*   batch = (const int*)  d_in[2];
    const float* W1    = (const float*)d_in[3];
    const float* b1    = (const float*)d_in[4];
    const float* W2    = (const float*)d_in[5];
    const float* b2    = (const float*)d_in[6];
    const float* fc1_w = (const float*)d_in[7];
    const float* fc1_b = (const float*)d_in[8];
    const float* fc2_w = (const float*)d_in[9];
    const float* fc2_b = (const float*)d_in[10];
    float* out = (float*)d_out;

    const int* src = eidx;
    const int* dst = eidx + N_EDGES;

    float* buf0 = (float*)d_ws;                       // lin output   (12.8 MB)
    float* buf1 = buf0 + (size_t)N_NODES * H;         // aggregator   (12.8 MB)
    float* pool = buf1 + (size_t)N_NODES * H;         // [64 x 64]
    float* cnt  = pool + N_GRAPHS * H;                // [64]

    const int MT = N_NODES / 16;                      // 3125 row tiles (exact)
    const int EW = (N_EDGES + 7) / 8;                 // 8 edge-waves / 256-thr block
    const int NF = (N_NODES * H) / 256;               // 12500 (exact)

    // Layer 1: lin = x @ W1 ; agg = b1 + scatter_add(lin[src] -> dst)
    gemm16_wmma<IN_CH, false><<<MT, 128, 0, stream>>>(x, W1, buf0);
    init_bias<<<NF, 256, 0, stream>>>(buf1, b1);
    edge_scatter<<<EW, 256, 0, stream>>>(buf0, src, dst, buf1);

    // Layer 2: lin = relu(agg) @ W2 (relu fused into A-load); agg = b2 + scatter
    gemm16_wmma<H, true><<<MT, 128, 0, stream>>>(buf1, W2, buf0);
    init_bias<<<NF, 256, 0, stream>>>(buf1, b2);
    edge_scatter<<<EW, 256, 0, stream>>>(buf0, src, dst, buf1);

    // Global mean pool over graphs (relu fused), then MLP head + log_softmax.
    zero_pool<<<(N_GRAPHS * H + 255) / 256, 256, 0, stream>>>(pool, cnt);
    pool_scatter<<<NF, 256, 0, stream>>>(buf1, batch, pool, cnt);
    head_kernel<<<1, 64, 0, stream>>>(pool, cnt, fc1_w, fc1_b, fc2_w, fc2_b, out);
}